// MSECG_36816459661403
// MI455X (gfx1250) — compile-verified
//
#include <hip/hip_runtime.h>
#include <hip/hip_bf16.h>

// ---------------- problem constants ----------------
#define Bb   8
#define CIN  12
#define LL   4096
#define Dd   64
#define DI   128
#define DS   16
#define DC   4
#define DTR  4
#define RR   10
#define KK   15
#define NTOK (Bb*LL)          // 32768 tokens
#define NBN  128              // BN partial blocks
#define LDSP 68               // padded LDS row stride (words); 68%64=4 -> conflict-free b64

typedef float v2f __attribute__((ext_vector_type(2)));
typedef float v8f __attribute__((ext_vector_type(8)));

// fp32 WMMA: D(16x16) += A(16x4) x B(4x16), all f32 (exact-precision matrix pipe)
__device__ __forceinline__ v8f wmma4(v2f a, v2f b, v8f c) {
  return __builtin_amdgcn_wmma_f32_16x16x4_f32(false, a, false, b, (short)0, c, false, false);
}

// ---------------- 1) conv_in + leaky_relu : (B,12,L) -> token-major (NTOK,64) ----
__global__ __launch_bounds__(256) void k_convin(const float* __restrict__ x,
                                                const float* __restrict__ W,
                                                const float* __restrict__ bias,
                                                float* __restrict__ xin) {
  int t = blockIdx.x * 256 + threadIdx.x;     // NTOK*64 threads
  int d = t & 63;
  int n = t >> 6;
  int b = n >> 12, l = n & (LL - 1);
  const float* xb = x + (size_t)b * CIN * LL;
  const float* wd = W + (size_t)d * CIN * KK;
  float acc = bias[d];
  #pragma unroll
  for (int c = 0; c < CIN; ++c) {
    #pragma unroll
    for (int k = 0; k < KK; ++k) {
      int lp = l + k - 7;
      float m = (lp >= 0 && lp < LL) ? 1.f : 0.f;
      int lc = lp < 0 ? 0 : (lp > LL - 1 ? LL - 1 : lp);
      acc += m * wd[c * KK + k] * xb[c * LL + lc];
    }
  }
  xin[(size_t)n * Dd + d] = acc >= 0.f ? acc : 0.01f * acc;
}

// ---- 2) in_proj GEMM (WMMA): wave = 1 n-tile x 4 m-tiles, B fragment reused 4x ----
__global__ __launch_bounds__(256) void k_inproj(const float* __restrict__ xin,
                                                const float* __restrict__ W,
                                                float* __restrict__ xs_raw,
                                                float* __restrict__ z) {
  int wave = (blockIdx.x * 256 + threadIdx.x) >> 5;   // 8192 waves
  int lane = threadIdx.x & 31;
  int half = lane >> 4, lid = lane & 15;
  int mg = wave & 3;             // m-group: tiles mg*4 .. mg*4+3 (of 16)
  int nT = wave >> 2;            // 2048 n-tiles
  int n0 = nT * 16;
  const float* Xb = xin + (size_t)n0 * 64;
  v8f acc[4] = {};
  #pragma unroll
  for (int k = 0; k < 64; k += 4) {
    int kb = k + half * 2;
    v2f bv; bv.x = Xb[lid * 64 + kb]; bv.y = Xb[lid * 64 + kb + 1];
    #pragma unroll
    for (int i = 0; i < 4; ++i) {
      const float* Wb = W + (size_t)(mg * 64 + i * 16 + lid) * 64;
      v2f a; a.x = Wb[kb]; a.y = Wb[kb + 1];
      acc[i] = wmma4(a, bv, acc[i]);
    }
  }
  int n = n0 + lid;
  float* dstbuf = (mg < 2) ? xs_raw : z;
  int mbase = (mg < 2) ? mg * 64 : (mg - 2) * 64;
  #pragma unroll
  for (int i = 0; i < 4; ++i) {
    float* dst = dstbuf + (size_t)n * DI + mbase + i * 16 + half * 8;
    #pragma unroll
    for (int r = 0; r < 8; ++r) dst[r] = acc[i][r];
  }
}

// ---------------- 3) causal depthwise conv (DC=4) + SiLU ----------------
__global__ __launch_bounds__(256) void k_dwconv(const float* __restrict__ xs_raw,
                                                const float* __restrict__ W,
                                                const float* __restrict__ bias,
                                                float* __restrict__ xs) {
  int t = blockIdx.x * 256 + threadIdx.x;    // NTOK*128
  int d = t & 127;
  int n = t >> 7;
  int l = n & (LL - 1);
  float acc = bias[d];
  #pragma unroll
  for (int j = 0; j < DC; ++j) {
    int off = j - (DC - 1);                  // -3..0
    float m = (l + off >= 0) ? 1.f : 0.f;
    int lp = (l + off >= 0) ? off : -l;      // clamp to l=0 row of same batch
    acc += m * W[d * DC + j] * xs_raw[((size_t)n + lp) * DI + d];
  }
  xs[(size_t)n * DI + d] = acc / (1.f + expf(-acc));
}

// ---------------- 4) x_proj: (NTOK,128) x (36,128)^T -> dbl ----------------
__global__ __launch_bounds__(256) void k_xproj(const float* __restrict__ xs,
                                               const float* __restrict__ W,
                                               float* __restrict__ dbl) {
  int t = blockIdx.x * 256 + threadIdx.x;    // NTOK*36
  int e = t % 36;
  int n = t / 36;
  const float* xr = xs + (size_t)n * DI;
  const float* wr = W + (size_t)e * DI;
  float acc = 0.f;
  #pragma unroll 8
  for (int k = 0; k < DI; ++k) acc += xr[k] * wr[k];
  dbl[(size_t)n * 36 + e] = acc;
}

// ---------------- 5) dt_proj + softplus ----------------
__global__ __launch_bounds__(256) void k_dtproj(const float* __restrict__ dbl,
                                                const float* __restrict__ W,
                                                const float* __restrict__ bias,
                                                float* __restrict__ dt) {
  int t = blockIdx.x * 256 + threadIdx.x;    // NTOK*128
  int d = t & 127;
  int n = t >> 7;
  float acc = bias[d];
  #pragma unroll
  for (int r = 0; r < DTR; ++r) acc += dbl[(size_t)n * 36 + r] * W[d * DTR + r];
  float sp = acc > 0.f ? acc + log1pf(expf(-acc)) : log1pf(expf(acc));
  dt[(size_t)n * DI + d] = sp;
}

// ------ 6) selective scan, fused D-skip + SiLU gate. lane=(state s), 16-lane reduce ----
__global__ __launch_bounds__(256) void k_scan(const float* __restrict__ dt,
                                              const float* __restrict__ xs,
                                              const float* __restrict__ dbl,
                                              const float* __restrict__ z,
                                              const float* __restrict__ A_log,
                                              const float* __restrict__ D_ssm,
                                              float* __restrict__ yact) {
  int t = blockIdx.x * 256 + threadIdx.x;    // 16384 threads = B*DI*DS
  int s = t & 15;
  int p = t >> 4;                            // (b,d) pair 0..1023
  int d = p & 127;
  int b = p >> 7;
  float A_ds = -expf(A_log[d * DS + s]);
  float Dskip = D_ssm[d];
  float h = 0.f;
  const size_t nb = (size_t)b << 12;
  for (int l = 0; l < LL; ++l) {
    size_t n = nb + l;
    float dtv = dt[n * DI + d];
    float xsv = xs[n * DI + d];
    float Bv = dbl[n * 36 + DTR + s];
    float Cv = dbl[n * 36 + DTR + DS + s];
    h = expf(dtv * A_ds) * h + dtv * Bv * xsv;
    float part = h * Cv;
    part += __shfl_xor(part, 1, 16);
    part += __shfl_xor(part, 2, 16);
    part += __shfl_xor(part, 4, 16);
    part += __shfl_xor(part, 8, 16);
    if (s == 0) {
      float y = part + xsv * Dskip;
      float zv = z[n * DI + d];
      yact[n * DI + d] = y * (zv / (1.f + expf(-zv)));
    }
  }
}

// ---- 7) out_proj GEMM (WMMA) + residual; wave = 1 n-tile x all 4 m-tiles ----
__global__ __launch_bounds__(256) void k_outproj(const float* __restrict__ yact,
                                                 const float* __restrict__ W,
                                                 const float* __restrict__ xin,
                                                 float* __restrict__ xm) {
  int wave = (blockIdx.x * 256 + threadIdx.x) >> 5;   // 2048 waves = n-tiles
  int lane = threadIdx.x & 31;
  int half = lane >> 4, lid = lane & 15;
  int n0 = wave * 16;
  const float* Xb = yact + (size_t)n0 * DI;
  v8f acc[4] = {};
  #pragma unroll
  for (int k = 0; k < DI; k += 4) {
    int kb = k + half * 2;
    v2f bv; bv.x = Xb[lid * DI + kb]; bv.y = Xb[lid * DI + kb + 1];
    #pragma unroll
    for (int i = 0; i < 4; ++i) {
      const float* Wb = W + (size_t)(i * 16 + lid) * DI;
      v2f a; a.x = Wb[kb]; a.y = Wb[kb + 1];
      acc[i] = wmma4(a, bv, acc[i]);
    }
  }
  int n = n0 + lid;
  #pragma unroll
  for (int i = 0; i < 4; ++i) {
    size_t base = (size_t)n * Dd + i * 16 + half * 8;
    #pragma unroll
    for (int r = 0; r < 8; ++r) xm[base + r] = acc[i][r] + xin[base + r];
  }
}

// -- 8) conv_merge implicit GEMM: LDS-staged 128-pos strip + halo, 4 m-tiles/wave --
__global__ __launch_bounds__(256) void k_convmerge(const float* __restrict__ xm,
                                                   const float* __restrict__ W,
                                                   const float* __restrict__ bias,
                                                   float* __restrict__ out) {
  __shared__ float smem[144 * LDSP];          // 128+14 rows (+2 pad) x 64 ch, padded stride
  int b = blockIdx.x >> 5;                    // 256 blocks = 8 b x 32 strips
  int l0 = (blockIdx.x & 31) << 7;            // strip * 128
  #pragma unroll
  for (int it = 0; it < 9; ++it) {            // 144*16 float4 slots / 256 threads
    int s = it * 256 + threadIdx.x;
    int r = s >> 4, c4 = (s & 15) << 2;
    int l = l0 - 7 + r;
    float4 v = make_float4(0.f, 0.f, 0.f, 0.f);
    if (l >= 0 && l < LL) v = *(const float4*)(xm + ((size_t)(b << 12) + l) * Dd + c4);
    *(float4*)(smem + r * LDSP + c4) = v;
  }
  __syncthreads();
  int wave = threadIdx.x >> 5;                // n-subtile 0..7
  int lane = threadIdx.x & 31;
  int half = lane >> 4, lid = lane & 15;
  v8f acc[4] = {};
  for (int k = 0; k < KK; ++k) {
    const float* brow = smem + (wave * 16 + lid + k) * LDSP;
    #pragma unroll
    for (int c = 0; c < Dd; c += 4) {
      int cb = c + half * 2;
      v2f bv; bv.x = brow[cb]; bv.y = brow[cb + 1];
      #pragma unroll
      for (int i = 0; i < 4; ++i) {
        const float* Wr = W + (size_t)(i * 16 + lid) * (Dd * KK) + k;
        v2f a; a.x = Wr[cb * KK]; a.y = Wr[(cb + 1) * KK];
        acc[i] = wmma4(a, bv, acc[i]);
      }
    }
  }
  int n = (b << 12) + l0 + wave * 16 + lid;
  #pragma unroll
  for (int i = 0; i < 4; ++i) {
    int mch = i * 16 + half * 8;
    float* dst = out + (size_t)n * Dd + mch;
    #pragma unroll
    for (int r2 = 0; r2 < 8; ++r2) dst[r2] = acc[i][r2] + bias[mch + r2];
  }
}

// ---------------- 9) BN partial sums (deterministic fixed-order, no atomics) -------
__global__ __launch_bounds__(256) void k_bnpart(const float* __restrict__ xmerge,
                                                float* __restrict__ part) {
  __shared__ float ssum[256], ssq[256];
  int c = threadIdx.x & 63, g = threadIdx.x >> 6;
  int tok0 = blockIdx.x * (NTOK / NBN);
  float s = 0.f, q = 0.f;
  for (int i = g; i < NTOK / NBN; i += 4) {
    float v = xmerge[(size_t)(tok0 + i) * Dd + c];
    s += v; q += v * v;
  }
  ssum[threadIdx.x] = s; ssq[threadIdx.x] = q;
  __syncthreads();
  if (g == 0) {
    #pragma unroll
    for (int gg = 1; gg < 4; ++gg) { s += ssum[gg * 64 + c]; q += ssq[gg * 64 + c]; }
    part[blockIdx.x * 64 + c] = s;
    part[NBN * 64 + blockIdx.x * 64 + c] = q;
  }
}

// ---------------- 10) BN finalize -> scale/shift ----------------
__global__ void k_bnfin(const float* __restrict__ part,
                        const float* __restrict__ gamma,
                        const float* __restrict__ beta,
                        float* __restrict__ sc) {
  int c = threadIdx.x;
  if (c >= 64) return;
  float s = 0.f, q = 0.f;
  for (int p = 0; p < NBN; ++p) { s += part[p * 64 + c]; q += part[NBN * 64 + p * 64 + c]; }
  const float inv = 1.f / (float)NTOK;
  float mean = s * inv;
  float var = q * inv - mean * mean;
  float scale = gamma[c] * rsqrtf(var + 1e-5f);
  sc[c] = scale;
  sc[64 + c] = beta[c] - mean * scale;
}

// ---------------- 11) BN apply + residual (in place) ----------------
__global__ __launch_bounds__(256) void k_bnapply(float* __restrict__ xmerge,
                                                 const float* __restrict__ sc,
                                                 const float* __restrict__ xin) {
  int t = blockIdx.x * 256 + threadIdx.x;    // NTOK*64
  int c = t & 63;
  xmerge[t] = xmerge[t] * sc[c] + sc[64 + c] + xin[t];
}

// -- 12) conv_ps1 implicit GEMM: LDS-staged 64-pos strip, M=120 padded to 128 --
__global__ __launch_bounds__(256) void k_convps1(const float* __restrict__ xmg,
                                                 const float* __restrict__ W,
                                                 const float* __restrict__ bias,
                                                 float* __restrict__ xps) {
  __shared__ float smem[80 * LDSP];           // 64+14 rows (+2 pad) x 64 ch
  int b = blockIdx.x >> 6;                    // 512 blocks = 8 b x 64 strips
  int l0 = (blockIdx.x & 63) << 6;            // strip * 64
  #pragma unroll
  for (int it = 0; it < 5; ++it) {            // 80*16 float4 slots / 256 threads
    int s = it * 256 + threadIdx.x;
    int r = s >> 4, c4 = (s & 15) << 2;
    int l = l0 - 7 + r;
    float4 v = make_float4(0.f, 0.f, 0.f, 0.f);
    if (l >= 0 && l < LL) v = *(const float4*)(xmg + ((size_t)(b << 12) + l) * Dd + c4);
    *(float4*)(smem + r * LDSP + c4) = v;
  }
  __syncthreads();
  int wave = threadIdx.x >> 5;
  int nsub = wave >> 1;                       // 0..3 (16 positions each)
  int mh = wave & 1;                          // m-tiles mh*4 .. mh*4+3
  int lane = threadIdx.x & 31;
  int half = lane >> 4, lid = lane & 15;
  v8f acc[4] = {};
  for (int k = 0; k < KK; ++k) {
    const float* brow = smem + (nsub * 16 + lid + k) * LDSP;
    #pragma unroll
    for (int c = 0; c < Dd; c += 4) {
      int cb = c + half * 2;
      v2f bv; bv.x = brow[cb]; bv.y = brow[cb + 1];
      #pragma unroll
      for (int i = 0; i < 4; ++i) {
        int wm = (mh * 4 + i) * 16 + lid;
        float wmask = (wm < RR * CIN) ? 1.f : 0.f;
        int wrow = wm < RR * CIN ? wm : RR * CIN - 1;
        const float* Wr = W + (size_t)wrow * (Dd * KK) + k;
        v2f a; a.x = wmask * Wr[cb * KK]; a.y = wmask * Wr[(cb + 1) * KK];
        acc[i] = wmma4(a, bv, acc[i]);
      }
    }
  }
  int n = (b << 12) + l0 + nsub * 16 + lid;
  #pragma unroll
  for (int i = 0; i < 4; ++i) {
    int mch = (mh * 4 + i) * 16 + half * 8;
    #pragma unroll
    for (int r2 = 0; r2 < 8; ++r2) {
      int m = mch + r2;
      if (m < RR * CIN) xps[(size_t)n * (RR * CIN) + m] = acc[i][r2] + bias[m];
    }
  }
}

// ------- 13) pixel-shuffle (r=10) + linear-interp upsample of raw x, write output ---
__global__ __launch_bounds__(256) void k_final(const float* __restrict__ xps,
                                               const float* __restrict__ x,
                                               float* __restrict__ out) {
  int t = blockIdx.x * 256 + threadIdx.x;    // B*12*L*R = 3,932,160
  int lo = t % (LL * RR);
  int rem = t / (LL * RR);
  int cout = rem % CIN;
  int b = rem / CIN;
  int l = lo / RR;
  int r = lo - l * RR;
  float ps = xps[((size_t)(b << 12) + l) * (RR * CIN) + r * CIN + cout];
  float src = fmaxf(((float)lo + 0.5f) * (1.f / RR) - 0.5f, 0.f);
  int i0 = (int)src; if (i0 > LL - 1) i0 = LL - 1;
  int i1 = i0 + 1;   if (i1 > LL - 1) i1 = LL - 1;
  float frac = src - (float)i0;
  const float* xr = x + (size_t)b * CIN * LL + (size_t)cout * LL;
  float up = xr[i0] * (1.f - frac) + xr[i1] * frac;
  out[t] = ps + up;
}

// =======================================================================
extern "C" void kernel_launch(void* const* d_in, const int* in_sizes, int n_in,
                              void* d_out, int out_size, void* d_ws, size_t ws_size,
                              hipStream_t stream) {
  (void)in_sizes; (void)n_in; (void)out_size; (void)ws_size;
  const float* x           = (const float*)d_in[0];
  const float* conv_in_w   = (const float*)d_in[1];
  const float* conv_in_b   = (const float*)d_in[2];
  const float* in_proj_w   = (const float*)d_in[3];
  const float* dw_conv_w   = (const float*)d_in[4];
  const float* dw_conv_b   = (const float*)d_in[5];
  const float* x_proj_w    = (const float*)d_in[6];
  const float* dt_proj_w   = (const float*)d_in[7];
  const float* dt_proj_b   = (const float*)d_in[8];
  const float* A_log       = (const float*)d_in[9];
  const float* D_ssm       = (const float*)d_in[10];
  const float* out_proj_w  = (const float*)d_in[11];
  const float* conv_merge_w= (const float*)d_in[12];
  const float* conv_merge_b= (const float*)d_in[13];
  const float* bn_gamma    = (const float*)d_in[14];
  const float* bn_beta     = (const float*)d_in[15];
  const float* conv_ps1_w  = (const float*)d_in[16];
  const float* conv_ps1_b  = (const float*)d_in[17];

  float* ws   = (float*)d_ws;
  float* x_in = ws;                          // (NTOK,64)        2,097,152
  float* bufA = ws + 2097152;                // xs_raw -> yact   4,194,304
  float* bufB = ws + 6291456;                // z -> xm          4,194,304
  float* bufC = ws + 10485760;               // xs -> xps        4,194,304
  float* bufD = ws + 14680064;               // dt -> xmerge     4,194,304
  float* dbl  = ws + 18874368;               // (NTOK,36)        1,179,648
  float* bnp  = ws + 20054016;               // BN partials      16,384
  float* bnsc = ws + 20070400;               // scale/shift      128
  float* outp = (float*)d_out;

  k_convin   <<<(NTOK*Dd)/256,  256, 0, stream>>>(x, conv_in_w, conv_in_b, x_in);
  k_inproj   <<<1024,           256, 0, stream>>>(x_in, in_proj_w, bufA, bufB);
  k_dwconv   <<<(NTOK*DI)/256,  256, 0, stream>>>(bufA, dw_conv_w, dw_conv_b, bufC);
  k_xproj    <<<(NTOK*36)/256,  256, 0, stream>>>(bufC, x_proj_w, dbl);
  k_dtproj   <<<(NTOK*DI)/256,  256, 0, stream>>>(dbl, dt_proj_w, dt_proj_b, bufD);
  k_scan     <<<64,             256, 0, stream>>>(bufD, bufC, dbl, bufB, A_log, D_ssm, bufA);
  k_outproj  <<<256,            256, 0, stream>>>(bufA, out_proj_w, x_in, bufB);
  k_convmerge<<<256,            256, 0, stream>>>(bufB, conv_merge_w, conv_merge_b, bufD);
  k_bnpart   <<<NBN,            256, 0, stream>>>(bufD, bnp);
  k_bnfin    <<<1,              64,  0, stream>>>(bnp, bn_gamma, bn_beta, bnsc);
  k_bnapply  <<<(NTOK*Dd)/256,  256, 0, stream>>>(bufD, bnsc, x_in);
  k_convps1  <<<512,            256, 0, stream>>>(bufD, conv_ps1_w, conv_ps1_b, bufC);
  k_final    <<<(Bb*CIN*LL*RR)/256, 256, 0, stream>>>(bufC, x, outp);
}